// GCN2Net_80479097193052
// MI455X (gfx1250) — compile-verified
//
#include <hip/hip_runtime.h>
#include <cmath>

typedef __attribute__((ext_vector_type(2))) float v2f;
typedef __attribute__((ext_vector_type(8))) float v8f;

#define HID 128
#define IN_DIM 512
#define N_LAYERS 8
#define ALPHA_C 0.1f
#define LDS_STRIDE 132   // 128 + 4 pad to avoid bank conflicts

// ---------------------------------------------------------------- utilities
__global__ __launch_bounds__(256) void zero_kernel(float* __restrict__ p, size_t n) {
    size_t i = (size_t)blockIdx.x * blockDim.x + threadIdx.x;
    size_t stride = (size_t)gridDim.x * blockDim.x;
    for (; i < n; i += stride) p[i] = 0.0f;
}

__global__ __launch_bounds__(256) void deg_kernel(const int* __restrict__ dst,
                                                  float* __restrict__ deg, int nE) {
    int e = blockIdx.x * blockDim.x + threadIdx.x;
    if (e < nE) unsafeAtomicAdd(&deg[dst[e]], 1.0f);
}

__global__ __launch_bounds__(256) void dinv_kernel(float* __restrict__ deg, int n) {
    int i = blockIdx.x * blockDim.x + threadIdx.x;
    if (i < n) deg[i] = rsqrtf(deg[i] + 1.0f);   // +1 = self loop; always > 0
}

__global__ __launch_bounds__(256) void norm_kernel(const int* __restrict__ src,
                                                   const int* __restrict__ dst,
                                                   const float* __restrict__ dinv,
                                                   float* __restrict__ norm, int nE) {
    int e = blockIdx.x * blockDim.x + threadIdx.x;
    if (e < nE) norm[e] = dinv[src[e]] * dinv[dst[e]];
}

// ------------------------------------------------- initial projection (WMMA)
// X[N,512] @ Win[512,128] + bin -> x0 and x. One wave per 16x16 tile.
__global__ __launch_bounds__(256) void inproj_kernel(const float* __restrict__ X,
                                                     const float* __restrict__ Win,
                                                     const float* __restrict__ bin,
                                                     float* __restrict__ x0,
                                                     float* __restrict__ x) {
    const int tid  = threadIdx.x;
    const int lane = tid & 31;
    const int wave = tid >> 5;            // 0..7 => N-tile
    const int rowBase = blockIdx.x * 16;
    const int m16  = lane & 15;
    const int half = lane >> 4;           // 0/1
    const int nCol = wave * 16 + m16;     // 0..127

    const float* __restrict__ xr = X + (size_t)(rowBase + m16) * IN_DIM;
    v8f acc = {};
    for (int k = 0; k < IN_DIM; k += 4) {
        int kb = k + 2 * half;
        v2f a, b;
        a.x = xr[kb];
        a.y = xr[kb + 1];
        b.x = Win[(size_t)kb * HID + nCol];
        b.y = Win[(size_t)(kb + 1) * HID + nCol];
        acc = __builtin_amdgcn_wmma_f32_16x16x4_f32(false, a, false, b,
                                                    (short)0, acc, false, false);
    }
    float bias = bin[nCol];
#pragma unroll
    for (int v = 0; v < 8; ++v) {
        int m = v + 8 * half;
        size_t off = (size_t)(rowBase + m) * HID + nCol;
        float val = acc[v] + bias;
        x0[off] = val;
        x[off]  = val;
    }
}

// ---------------------------------------------------------- edge scatter SpMM
// agg[dst] += norm * x[src]; one wave per edge, 4 coalesced f32 atomics/lane.
__global__ __launch_bounds__(256) void scatter_kernel(const float* __restrict__ x,
                                                      const int* __restrict__ src,
                                                      const int* __restrict__ dst,
                                                      const float* __restrict__ norm,
                                                      float* __restrict__ agg, int nE) {
    int gwave = (int)((blockIdx.x * blockDim.x + threadIdx.x) >> 5);
    int lane  = threadIdx.x & 31;
    if (gwave >= nE) return;
    int s = src[gwave], d = dst[gwave];
    float w = norm[gwave];
    const float* __restrict__ xs = x + (size_t)s * HID;
    float* __restrict__ ad = agg + (size_t)d * HID;
#pragma unroll
    for (int j = 0; j < 4; ++j) {
        int f = lane + 32 * j;
        unsafeAtomicAdd(&ad[f], w * xs[f]);
    }
}

// --------------------------------------------- fused layer (residual + WMMA)
// h = 0.9*(agg + dinv^2 * x) + 0.1*x0   (staged in LDS)
// out = relu((1-beta)*h + beta * h @ W) ; out may alias agg.
__global__ __launch_bounds__(256) void layer_kernel(const float* __restrict__ agg,
                                                    const float* __restrict__ x,
                                                    const float* __restrict__ x0,
                                                    const float* __restrict__ dinv,
                                                    const float* __restrict__ W,
                                                    float* __restrict__ out,
                                                    float beta) {
    __shared__ float hsm[16 * LDS_STRIDE];
    const int tid = threadIdx.x;
    const int rowBase = blockIdx.x * 16;

    for (int idx = tid; idx < 16 * HID; idx += 256) {
        int m = idx >> 7, f = idx & (HID - 1);
        int row = rowBase + m;
        float di = dinv[row];
        size_t off = (size_t)row * HID + f;
        float h = (1.0f - ALPHA_C) * (agg[off] + di * di * x[off]) + ALPHA_C * x0[off];
        hsm[m * LDS_STRIDE + f] = h;
    }
    __syncthreads();

    const int lane = tid & 31;
    const int wave = tid >> 5;
    const int m16  = lane & 15;
    const int half = lane >> 4;
    const int nCol = wave * 16 + m16;

    v8f acc = {};
    for (int k = 0; k < HID; k += 4) {
        int kb = k + 2 * half;
        v2f a, b;
        a.x = hsm[m16 * LDS_STRIDE + kb];
        a.y = hsm[m16 * LDS_STRIDE + kb + 1];
        b.x = W[(size_t)kb * HID + nCol];
        b.y = W[(size_t)(kb + 1) * HID + nCol];
        acc = __builtin_amdgcn_wmma_f32_16x16x4_f32(false, a, false, b,
                                                    (short)0, acc, false, false);
    }
#pragma unroll
    for (int v = 0; v < 8; ++v) {
        int m = v + 8 * half;
        float h = hsm[m * LDS_STRIDE + nCol];
        float val = (1.0f - beta) * h + beta * acc[v];
        out[(size_t)(rowBase + m) * HID + nCol] = fmaxf(val, 0.0f);
    }
}

// -------------------------------------------------------- output projection
__global__ __launch_bounds__(256) void outproj_kernel(const float* __restrict__ x,
                                                      const float* __restrict__ Wout,
                                                      const float* __restrict__ bout,
                                                      float* __restrict__ out, int n) {
    int node = (int)((blockIdx.x * blockDim.x + threadIdx.x) >> 5);
    int lane = threadIdx.x & 31;
    if (node >= n) return;
    const float* __restrict__ xr = x + (size_t)node * HID;
    float s = 0.0f;
#pragma unroll
    for (int j = 0; j < 4; ++j) {
        int f = lane + 32 * j;
        s += xr[f] * Wout[f];
    }
#pragma unroll
    for (int off = 16; off > 0; off >>= 1) s += __shfl_down(s, off, 32);
    if (lane == 0) out[node] = s + bout[0];
}

// ---------------------------------------------------------------- launcher
static inline size_t align256(size_t x) { return (x + 255) & ~(size_t)255; }

extern "C" void kernel_launch(void* const* d_in, const int* in_sizes, int n_in,
                              void* d_out, int out_size, void* d_ws, size_t ws_size,
                              hipStream_t stream) {
    const float* X    = (const float*)d_in[0];
    const int*   ei   = (const int*)d_in[1];      // [2, E]
    // d_in[2] edge_weight: unused by reference (gcn_norm uses unit weights)
    const float* Win  = (const float*)d_in[3];    // [512,128]
    const float* bin  = (const float*)d_in[4];    // [128]
    const float* Wl   = (const float*)d_in[5];    // [8,128,128]
    const float* Wout = (const float*)d_in[6];    // [128,1]
    const float* bout = (const float*)d_in[7];    // [1]
    float* out = (float*)d_out;

    const int N = in_sizes[0] / IN_DIM;           // 100000
    const int E = in_sizes[1] / 2;                // 1600000
    const int* srcIdx = ei;
    const int* dstIdx = ei + E;

    // workspace carve-out
    char* w = (char*)d_ws;
    float* dinv = (float*)w;                 w += align256((size_t)N * 4);
    float* norm = (float*)w;                 w += align256((size_t)E * 4);
    float* x0   = (float*)w;                 w += align256((size_t)N * HID * 4);
    float* bufA = (float*)w;                 w += align256((size_t)N * HID * 4);
    float* bufB = (float*)w;                 w += align256((size_t)N * HID * 4);
    (void)ws_size; (void)n_in; (void)out_size;

    const size_t stateElems = (size_t)N * HID;

    // --- gcn_norm ---
    zero_kernel<<<1024, 256, 0, stream>>>(dinv, (size_t)N);      // deg accumulator
    deg_kernel<<<(E + 255) / 256, 256, 0, stream>>>(dstIdx, dinv, E);
    dinv_kernel<<<(N + 255) / 256, 256, 0, stream>>>(dinv, N);
    norm_kernel<<<(E + 255) / 256, 256, 0, stream>>>(srcIdx, dstIdx, dinv, norm, E);

    // --- initial projection (WMMA) ---
    inproj_kernel<<<N / 16, 256, 0, stream>>>(X, Win, bin, x0, bufA);

    // --- GCNII layers ---
    float* xBuf = bufA;
    float* aggBuf = bufB;
    for (int i = 0; i < N_LAYERS; ++i) {
        float beta = logf(0.5f / (float)(i + 1) + 1.0f);
        zero_kernel<<<4096, 256, 0, stream>>>(aggBuf, stateElems);
        scatter_kernel<<<(E + 7) / 8, 256, 0, stream>>>(xBuf, srcIdx, dstIdx, norm,
                                                        aggBuf, E);
        layer_kernel<<<N / 16, 256, 0, stream>>>(aggBuf, xBuf, x0, dinv,
                                                 Wl + (size_t)i * HID * HID,
                                                 aggBuf, beta);   // out aliases agg
        float* t = xBuf; xBuf = aggBuf; aggBuf = t;
    }

    // --- output projection ---
    outproj_kernel<<<(N + 7) / 8, 256, 0, stream>>>(xBuf, Wout, bout, out, N);
}